// M_Noise_Deepspeech_35003983462728
// MI455X (gfx1250) — compile-verified
//
#include <hip/hip_runtime.h>
#include <cstdint>

#define KBINS 161
#define TT 8          // time columns per block
#define NTHREADS 256

// Low 32 bits of a flat shared-aperture address are the wave-relative LDS
// byte offset (CDNA5 ISA per-aperture calc).
__device__ __forceinline__ uint32_t lds_off(const void* p) {
  return (uint32_t)(uintptr_t)p;
}

__global__ __launch_bounds__(NTHREADS)
void mnoise_smooth_kernel(const float* __restrict__ m,
                          const float* __restrict__ x,
                          float* __restrict__ out,
                          int T) {
  // Interleaved LDS tile, layout [i*TT + tt]: .x = |m| (am), .y = x/S (g)
  __shared__ float2 sWG[KBINS * TT];
  __shared__ int    sMaxD;

  const int t0  = blockIdx.x * TT;
  const int tid = threadIdx.x;
  if (tid == 0) sMaxD = 0;

  // ---- Phase 1: async-stage m and x (K x TT tile) straight into LDS ----
  for (int e = tid; e < KBINS * TT; e += NTHREADS) {
    const int i  = e / TT;
    const int tt = e - i * TT;
    const int t  = t0 + tt;
    if (t < T) {
      const float* pm = m + (size_t)i * T + t;
      const float* px = x + (size_t)i * T + t;
      const uint32_t la = lds_off(&sWG[e].x);
      const uint32_t lg = lds_off(&sWG[e].y);
      asm volatile("global_load_async_to_lds_b32 %0, %1, off"
                   :: "v"(la), "v"(pm) : "memory");
      asm volatile("global_load_async_to_lds_b32 %0, %1, off"
                   :: "v"(lg), "v"(px) : "memory");
    }
  }
  asm volatile("s_wait_asynccnt 0" ::: "memory");
  __syncthreads();

  // ---- Phase 2: closed-form window normalization (in place) ----
  // Normalization cancels the /am^2, so w_norm = relu(am-|j-i|)/S with S the
  // boundary-clipped triangle sum in closed form. am <= 1 reduces exactly to
  // the reference's delta branch: D=0, S=am, weight at j==i is am -> x.
  int myMaxD = 0;
  for (int e = tid; e < KBINS * TT; e += NTHREADS) {
    const int i  = e / TT;
    const int tt = e - i * TT;
    if (t0 + tt >= T) continue;
    float2 wg = sWG[e];                        // ds_load_b64
    float am = fmaxf(fabsf(wg.x), 1e-20f);
    float Df = ceilf(am) - 1.0f;               // max |j-i| with positive weight
    float nl = fminf(Df, (float)i);            // left taps inside [0, K)
    float nr = fminf(Df, (float)(KBINS - 1 - i));
    float S  = am * (1.0f + nl + nr)
             - 0.5f * (nl * (nl + 1.0f) + nr * (nr + 1.0f));
    wg.x = am;
    wg.y = wg.y / S;
    sWG[e] = wg;                               // ds_store_b64
    myMaxD = max(myMaxD, (int)fminf(Df, (float)(KBINS - 1)));
  }
  atomicMax(&sMaxD, myMaxD);   // ds_max on LDS
  __syncthreads();

  // ---- Phase 3: radius-trimmed gather  out[j,t] = sum_i relu(am_i-|j-i|)*g_i
  const int maxD = sMaxD;
  for (int o = tid; o < KBINS * TT; o += NTHREADS) {
    const int j  = o / TT;
    const int tt = o - j * TT;
    const int t  = t0 + tt;
    if (t >= T) continue;
    int ilo = j - maxD; if (ilo < 0) ilo = 0;
    int ihi = j + maxD; if (ihi > KBINS - 1) ihi = KBINS - 1;
    const float jf = (float)j;
    float acc = 0.0f;
    for (int i = ilo; i <= ihi; ++i) {
      const float2 wg = sWG[i * TT + tt];      // single ds_load_b64 per tap
      const float w = fmaxf(wg.x - fabsf(jf - (float)i), 0.0f);
      acc = fmaf(w, wg.y, acc);
    }
    out[(size_t)j * T + t] = acc;
  }
}

extern "C" void kernel_launch(void* const* d_in, const int* in_sizes, int n_in,
                              void* d_out, int out_size, void* d_ws, size_t ws_size,
                              hipStream_t stream) {
  const float* m = (const float*)d_in[0];   // (K, T, 1) fp32
  const float* x = (const float*)d_in[1];   // (1, 1, K, T) fp32
  float* out = (float*)d_out;               // (1, 1, K, T) fp32

  int T = (out_size > 0) ? (out_size / KBINS) : 2000;
  dim3 grid((T + TT - 1) / TT);
  mnoise_smooth_kernel<<<grid, NTHREADS, 0, stream>>>(m, x, out, T);
}